// ts_pct_14362370638100
// MI455X (gfx1250) — compile-verified
//
#include <hip/hip_runtime.h>

typedef float v2f __attribute__((ext_vector_type(2)));
typedef float v8f __attribute__((ext_vector_type(8)));

#define T_IN   2048
#define T_OUT  2033   // T - WINDOW + 1
#define NF     64
#define ROW    66
#define EPSV   1e-5f

// One wave computes a 16(t') x 16(f) output tile:
//   Out = G(16x32 band) x X(32x16)  via 8 chained V_WMMA_F32_16X16X4_F32
// done twice (num with x*indicator, den with |x|), then out = num/(den+eps).
//
// Bounds handling: loads are anchored at tload = min(tbase, T_IN-32), so every
// address is in-bounds and affine -> pure immediate-offset loads. The band is
// shifted by (tbase - tload) so stored outputs are exact; truncated windows
// only hit t' >= T_OUT, which are never stored. EXEC stays all-1s throughout.
__global__ __launch_bounds__(256) void win_norm_wmma(const float* __restrict__ in,
                                                     float* __restrict__ out) {
    const int wid   = blockIdx.x * 8 + (threadIdx.x >> 5);   // 65536 waves total
    const int lane  = threadIdx.x & 31;
    const int ftile = wid & 3;            // 4 feature tiles of 16
    const int ttile = (wid >> 2) & 127;   // 128 time tiles of 16 outputs
    const int b     = wid >> 9;           // 128 batches

    const int m = lane & 15;              // A-matrix row (= output t' within tile)
    const int h = lane >> 4;              // half-wave selector (K grouping)
    const int f = ftile * 16 + m;         // B/D column (feature); N == lane&15
    const int tbase = ttile * 16;

    // Clamped anchor: rows tload .. tload+31 are always in [0, T_IN).
    const int tload = (tbase <= T_IN - 32) ? tbase : (T_IN - 32);
    const int msh   = m + (tbase - tload);   // band shift: 0, or 16 on last tile

    // One base pointer per lane; all chunk loads are immediate offsets off it.
    const float* base = in + ((size_t)b * T_IN + (size_t)(tload + 2 * h)) * ROW;
    const float* px   = base + f;    // this lane's feature column
    const float* pi   = base + 64;   // indicator column (broadcast in half-wave)

    v8f cnum = {};
    v8f cden = {};

#pragma unroll
    for (int c = 0; c < 8; ++c) {
        // This lane supplies K-slices j0 = 4c+2h and j1 = j0+1 for A and B.
        const float x0 = px[c * 4 * ROW];
        const float i0 = pi[c * 4 * ROW];
        const float x1 = px[c * 4 * ROW + ROW];
        const float i1 = pi[c * 4 * ROW + ROW];

        const int j0 = 4 * c + 2 * h;

        // Shifted band: 1 iff (j - msh) in [0,16).
        v2f a;
        a.x = ((unsigned)(j0     - msh) < 16u) ? 1.0f : 0.0f;
        a.y = ((unsigned)(j0 + 1 - msh) < 16u) ? 1.0f : 0.0f;

        v2f bn, bd;
        bn.x = x0 * i0;             bn.y = x1 * i1;
        bd.x = __builtin_fabsf(x0); bd.y = __builtin_fabsf(x1);

        cnum = __builtin_amdgcn_wmma_f32_16x16x4_f32(false, a, false, bn,
                                                     (short)0, cnum, false, false);
        cden = __builtin_amdgcn_wmma_f32_16x16x4_f32(false, a, false, bd,
                                                     (short)0, cden, false, false);
    }

    // D layout: lane holds (M = v + 8h, N = lane&15) -> t' = tbase + 8h + v, feature f.
    const int tstart = tbase + 8 * h;
    float* op = out + ((size_t)b * NF + f) * T_OUT + tstart;

#pragma unroll
    for (int v = 0; v < 8; ++v) {
        if (tstart + v < T_OUT) {
            op[v] = cnum[v] / (cden[v] + EPSV);
        }
    }
}

extern "C" void kernel_launch(void* const* d_in, const int* in_sizes, int n_in,
                              void* d_out, int out_size, void* d_ws, size_t ws_size,
                              hipStream_t stream) {
    const float* in  = (const float*)d_in[0];
    float*       out = (float*)d_out;
    // 128 batches * 128 t-tiles * 4 f-tiles = 65536 waves; 8 waves / block.
    win_norm_wmma<<<dim3(8192), dim3(256), 0, stream>>>(in, out);
}